// FeatureTransformLayer_73761768341930
// MI455X (gfx1250) — compile-verified
//
#include <hip/hip_runtime.h>

typedef float v2f __attribute__((ext_vector_type(2)));
typedef float v4f __attribute__((ext_vector_type(4)));
typedef float v8f __attribute__((ext_vector_type(8)));

#define F_IN 64
#define OUT_W 512
#define LDS_STRIDE 516   // pad: breaks bank aliasing between rows m and m+8; keeps 16B alignment

// Hardware tanh (V_TANH_F32 trans op on gfx1250); branch-free fallback via exp2+rcp.
__device__ __forceinline__ float hw_tanh(float v) {
#if __has_builtin(__builtin_amdgcn_tanhf)
    return __builtin_amdgcn_tanhf(v);
#elif __has_builtin(__builtin_amdgcn_tanh_f32)
    return __builtin_amdgcn_tanh_f32(v);
#else
    // tanh(x) = sign(x) * (1 - 2 / (exp2(2|x|*log2e) + 1)); overflow -> exp2=inf -> rcp=0 -> 1
    float ax = __builtin_fabsf(v);
    float e  = __builtin_amdgcn_exp2f(ax * 2.8853900817779268f); // 2*log2(e)
    float r  = __builtin_amdgcn_rcpf(e + 1.0f);
    float t  = __builtin_fmaf(-2.0f, r, 1.0f);
    return __builtin_copysignf(t, v);
#endif
}

__global__ __launch_bounds__(256)
void ftl_wmma_kernel(const float* __restrict__ x,
                     const float* __restrict__ W,     // (4,16,128)
                     const float* __restrict__ bias,  // (4,128)
                     const int*   __restrict__ idx,   // (4,16) flat
                     float* __restrict__ out,         // (N,512)
                     int n_rows)
{
    __shared__ __align__(16) float stage[16 * LDS_STRIDE];
    __shared__ int s_idx[64];

    const int tid   = threadIdx.x;
    const int wave  = tid >> 5;
    const int lane  = tid & 31;
    const int nloc  = lane & 15;   // M (for A) or N (for B/C) within half
    const int khalf = lane >> 4;   // 0 -> K low pair, 1 -> K high pair

    if (tid < 64) s_idx[tid] = idx[tid];
    __syncthreads();

    const int colBase  = wave * 64;      // global output column base for this wave
    const int blk      = wave >> 1;      // feature block 0..3 (64 cols sit inside one 128-col block)
    const int kbase    = blk * 16;       // idx table base for this block

    // ---- preload B operands (W tiles) into registers: 4 col-tiles x 4 K-chunks ----
    // B layout (4x16, 2 VGPRs): v.x = B[k][n], v.y = B[k+1][n], k = q*4 + khalf*2, n = lane%16
    v2f   Bt[4][4];
    float bv[4];
    const float* Wb = W + blk * (16 * 128);
    #pragma unroll
    for (int t = 0; t < 4; ++t) {
        const int oc = colBase - blk * 128 + t * 16 + nloc;   // column within block, 0..127
        bv[t] = bias[blk * 128 + oc];
        #pragma unroll
        for (int q = 0; q < 4; ++q) {
            const int k = q * 4 + khalf * 2;
            Bt[t][q].x = Wb[(k    ) * 128 + oc];
            Bt[t][q].y = Wb[(k + 1) * 128 + oc];
        }
    }

    const int row0 = blockIdx.x * 16;

    // ---- load A: 16x16 slice of x for this feature block, as 4 chunks of K=4 ----
    // A layout (16x4, 2 VGPRs): v.x = A[m][k], v.y = A[m][k+1], m = lane%16, k = q*4 + khalf*2
    int arow = row0 + nloc;
    if (arow >= n_rows) arow = n_rows - 1;           // clamp (N%16==0 anyway; avoids OOB reads)
    const float* xr = x + (long)arow * F_IN;
    v2f A[4];
    #pragma unroll
    for (int q = 0; q < 4; ++q) {
        const int k = q * 4 + khalf * 2;
        A[q].x = xr[s_idx[kbase + k]];
        A[q].y = xr[s_idx[kbase + k + 1]];
    }

    // ---- accumulate: C seeded with bias (C[m][n] = bias[n] for all m) ----
    v8f C[4];
    #pragma unroll
    for (int t = 0; t < 4; ++t)
        #pragma unroll
        for (int r = 0; r < 8; ++r) C[t][r] = bv[t];

    #pragma unroll
    for (int q = 0; q < 4; ++q) {
        #pragma unroll
        for (int t = 0; t < 4; ++t) {
            C[t] = __builtin_amdgcn_wmma_f32_16x16x4_f32(
                /*neg_a=*/false, A[q], /*neg_b=*/false, Bt[t][q],
                /*c_mod=*/(short)0, C[t], /*reuse_a=*/false, /*reuse_b=*/false);
        }
    }

    // ---- tanh (single V_TANH_F32 per element) + stage to LDS ----
    // C/D layout: VGPR r, lane half -> M = khalf*8 + r, N = lane%16
    #pragma unroll
    for (int t = 0; t < 4; ++t) {
        const int col = colBase + t * 16 + nloc;
        #pragma unroll
        for (int r = 0; r < 8; ++r) {
            const int m = khalf * 8 + r;
            stage[m * LDS_STRIDE + col] = hw_tanh(C[t][r]);
        }
    }
    __syncthreads();

    // ---- cooperative wide non-temporal stores: 16x512 tile, 8 float4 per thread ----
    #pragma unroll
    for (int chunk = 0; chunk < 8; ++chunk) {
        const int f = chunk * 1024 + tid * 4;     // flat index into 16x512 tile
        const int r = f >> 9;                      // row 0..15
        const int c = f & 511;                     // col 0..511
        v4f v = *(const v4f*)&stage[r * LDS_STRIDE + c];
        const long grow = (long)row0 + r;
        if (grow < n_rows) {
            __builtin_nontemporal_store(v, (v4f*)(out + grow * OUT_W + c));
        }
    }
}

extern "C" void kernel_launch(void* const* d_in, const int* in_sizes, int n_in,
                              void* d_out, int out_size, void* d_ws, size_t ws_size,
                              hipStream_t stream) {
    const float* x    = (const float*)d_in[0];   // (N, 64)
    const float* W    = (const float*)d_in[1];   // (4, 16, 128)
    const float* bias = (const float*)d_in[2];   // (4, 128)
    const int*   idx  = (const int*)d_in[3];     // (4, 16)
    float* out = (float*)d_out;                  // (N, 512)

    const int n_rows = in_sizes[0] / F_IN;
    const int grid   = (n_rows + 15) / 16;       // 6250 for N=100000

    hipLaunchKernelGGL(ftl_wmma_kernel, dim3(grid), dim3(256), 0, stream,
                       x, W, bias, idx, out, n_rows);
}